// ResidualGraphConv_53558242181211
// MI455X (gfx1250) — compile-verified
//
#include <hip/hip_runtime.h>
#include <hip/hip_bf16.h>

// ---------------------------------------------------------------------------
// ResidualGraphConv on MI455X (gfx1250, wave32).
// Layout strategy:
//   features / final output : channel-major (B, C, N)  [reference layout]
//   GEMM outputs, edge data : node-major   (B, N, C)   [coalesced edge ops]
// GEMMs use V_WMMA_F32_16X16X4_F32 (fp32 matrix path, K-loop of 32 per tile).
// ---------------------------------------------------------------------------

typedef __attribute__((ext_vector_type(2))) float v2f;
typedef __attribute__((ext_vector_type(8))) float v8f;

#define C_DIM 128
#define EPS 1e-5f

// --- transpose the 128x128 weights once: Wt[c][o] = W[o][c] ----------------
__global__ void transpose_w(const float* __restrict__ W0, const float* __restrict__ W1,
                            const float* __restrict__ W2, const float* __restrict__ W3,
                            float* __restrict__ Wt) {
    const float* W;
    if      (blockIdx.x == 0) W = W0;
    else if (blockIdx.x == 1) W = W1;
    else if (blockIdx.x == 2) W = W2;
    else                      W = W3;
    float* o = Wt + (size_t)blockIdx.x * C_DIM * C_DIM;
    for (int i = threadIdx.x; i < C_DIM * C_DIM; i += blockDim.x) {
        int r = i >> 7, c = i & 127;          // W[r][c]
        o[c * C_DIM + r] = W[i];              // Wt[c][r]
    }
}

// --- dual GEMM: outS = X^T * WtS + bS ; outO = X^T * WtO + bO --------------
// X  : (B, C, N) channel-major, ld = N
// out: (B, N, C) node-major
// Each wave: 16 nodes x 128 outs for BOTH weight matrices (A reused).
__global__ void __launch_bounds__(128)
gemm_dual(const float* __restrict__ X,
          const float* __restrict__ WtS, const float* __restrict__ WtO,
          const float* __restrict__ bS,  const float* __restrict__ bO,
          float* __restrict__ outS, float* __restrict__ outO, int N) {
    const int lane = threadIdx.x & 31;
    const int wave = threadIdx.x >> 5;
    const int b    = blockIdx.y;
    const int n0   = (blockIdx.x * 4 + wave) * 16;
    if (n0 + 16 > N) return;                  // N is a multiple of 16 here

    const float* Xb = X + (size_t)b * C_DIM * N;
    const int m  = lane & 15;                 // node within A tile / out col
    const int hi = lane >> 4;                 // half-wave select

    v8f accS[8], accO[8];
#pragma unroll
    for (int j = 0; j < 8; ++j) {
        accS[j] = (v8f){0.f,0.f,0.f,0.f,0.f,0.f,0.f,0.f};
        accO[j] = (v8f){0.f,0.f,0.f,0.f,0.f,0.f,0.f,0.f};
    }

    for (int kk = 0; kk < C_DIM; kk += 4) {
        // A fragment (16 nodes x 4 channels), X^T tile:
        //   VGPR0: lanes0-15 K=kk, lanes16-31 K=kk+2 ; VGPR1: K=kk+1 / kk+3
        const int ka = kk + 2 * hi;
        v2f a;
        a.x = Xb[(size_t)ka       * N + n0 + m];
        a.y = Xb[(size_t)(ka + 1) * N + n0 + m];
#pragma unroll
        for (int jt = 0; jt < 8; ++jt) {
            const int col = jt * 16 + m;
            // B fragment (4 channels x 16 outs) from Wt (C x O row-major):
            //   VGPR0: lanes0-15 K=kk, lanes16-31 K=kk+1 ; VGPR1: K=kk+2 / kk+3
            v2f bs, bo;
            bs.x = WtS[(kk + hi)     * C_DIM + col];
            bs.y = WtS[(kk + 2 + hi) * C_DIM + col];
            bo.x = WtO[(kk + hi)     * C_DIM + col];
            bo.y = WtO[(kk + 2 + hi) * C_DIM + col];
            accS[jt] = __builtin_amdgcn_wmma_f32_16x16x4_f32(
                false, a, false, bs, (short)0, accS[jt], false, false);
            accO[jt] = __builtin_amdgcn_wmma_f32_16x16x4_f32(
                false, a, false, bo, (short)0, accO[jt], false, false);
        }
    }

    // D tile layout: VGPR r -> M = r + 8*hi, N(col) = lane&15
#pragma unroll
    for (int jt = 0; jt < 8; ++jt) {
        const int col = jt * 16 + m;
        const float biasS = bS[col];
        const float biasO = bO[col];
#pragma unroll
        for (int r = 0; r < 8; ++r) {
            const int row = n0 + r + 8 * hi;
            const size_t o = ((size_t)b * N + row) * C_DIM + col;
            outS[o] = accS[jt][r] + biasS;
            outO[o] = accO[jt][r] + biasO;
        }
    }
}

// --- edge scatter-add: h[b, r[e], :] += fo[b, g[e], :] ---------------------
__global__ void __launch_bounds__(128)
scatter_add(float* __restrict__ h, const float* __restrict__ fo,
            const int* __restrict__ ri, const int* __restrict__ gi, int N) {
    const int e = blockIdx.x;
    const int b = blockIdx.y;
    const int c = threadIdx.x;
    const int r = ri[e];
    const int g = gi[e];
    atomicAdd(&h[((size_t)b * N + r) * C_DIM + c],
              fo[((size_t)b * N + g) * C_DIM + c]);
}

// --- instance-norm stats, node-major input (B,N,C) -------------------------
__global__ void __launch_bounds__(128)
stats_nmajor(const float* __restrict__ h, float* __restrict__ sumbuf, int N, int npb) {
    const int b  = blockIdx.y;
    const int c  = threadIdx.x;
    const int n0 = blockIdx.x * npb;
    const int n1 = min(n0 + npb, N);
    float s = 0.f, q = 0.f;
    for (int n = n0; n < n1; ++n) {
        float v = h[((size_t)b * N + n) * C_DIM + c];
        s += v; q += v * v;
    }
    atomicAdd(&sumbuf[(b * C_DIM + c) * 2 + 0], s);
    atomicAdd(&sumbuf[(b * C_DIM + c) * 2 + 1], q);
}

__global__ void stats_finalize(const float* __restrict__ sumbuf,
                               float* __restrict__ stats, int N) {
    const int i = threadIdx.x;                // 256 = B*C
    const float inv = 1.f / (float)N;
    float m = sumbuf[i * 2 + 0] * inv;
    float v = sumbuf[i * 2 + 1] * inv - m * m;
    stats[i * 2 + 0] = m;
    stats[i * 2 + 1] = rsqrtf(v + EPS);
}

// --- instance-norm stats, channel-major input (B,C,N) ----------------------
__global__ void __launch_bounds__(256)
stats_cmajor(const float* __restrict__ x, float* __restrict__ stats, int N) {
    __shared__ float s1[256];
    __shared__ float s2[256];
    const int b = blockIdx.y, c = blockIdx.x;
    const float* p = x + ((size_t)b * C_DIM + c) * N;
    float s = 0.f, q = 0.f;
    for (int n = threadIdx.x; n < N; n += 256) {
        float v = p[n];
        s += v; q += v * v;
    }
    s1[threadIdx.x] = s; s2[threadIdx.x] = q;
    __syncthreads();
    for (int o = 128; o > 0; o >>= 1) {
        if ((int)threadIdx.x < o) {
            s1[threadIdx.x] += s1[threadIdx.x + o];
            s2[threadIdx.x] += s2[threadIdx.x + o];
        }
        __syncthreads();
    }
    if (threadIdx.x == 0) {
        float m = s1[0] / (float)N;
        float v = s2[0] / (float)N - m * m;
        stats[(b * C_DIM + c) * 2 + 0] = m;
        stats[(b * C_DIM + c) * 2 + 1] = rsqrtf(v + EPS);
    }
}

// --- normalize + prelu + transpose (B,N,C) -> (B,C,N) ----------------------
__global__ void __launch_bounds__(256)
norm_transpose_prelu(const float* __restrict__ h, const float* __restrict__ stats,
                     const float* __restrict__ alpha, float* __restrict__ xcm, int N) {
    __shared__ float tile[32][129];
    const int b  = blockIdx.y;
    const int n0 = blockIdx.x * 32;
    const float a = *alpha;
#pragma unroll
    for (int it = 0; it < 16; ++it) {
        int idx = threadIdx.x + it * 256;
        int nn = idx >> 7, c = idx & 127;
        int n = n0 + nn;
        float v = 0.f;
        if (n < N) {
            v = h[((size_t)b * N + n) * C_DIM + c];
            v = (v - stats[(b * C_DIM + c) * 2]) * stats[(b * C_DIM + c) * 2 + 1];
            v = v >= 0.f ? v : a * v;
        }
        tile[nn][c] = v;
    }
    __syncthreads();
#pragma unroll
    for (int it = 0; it < 16; ++it) {
        int idx = threadIdx.x + it * 256;
        int c = idx >> 5, nn = idx & 31;
        int n = n0 + nn;
        if (n < N) xcm[((size_t)b * C_DIM + c) * N + n] = tile[nn][c];
    }
}

// --- final: prelu( norm(h2) + norm(features) ) -> (B,C,N) ------------------
__global__ void __launch_bounds__(256)
final_fuse(const float* __restrict__ h2, const float* __restrict__ stats2,
           const float* __restrict__ x,  const float* __restrict__ statsF,
           const float* __restrict__ alpha, float* __restrict__ out, int N) {
    __shared__ float tile[32][129];
    const int b  = blockIdx.y;
    const int n0 = blockIdx.x * 32;
    const float a = *alpha;
#pragma unroll
    for (int it = 0; it < 16; ++it) {
        int idx = threadIdx.x + it * 256;
        int nn = idx >> 7, c = idx & 127;
        int n = n0 + nn;
        float v = 0.f;
        if (n < N) {
            v = h2[((size_t)b * N + n) * C_DIM + c];
            v = (v - stats2[(b * C_DIM + c) * 2]) * stats2[(b * C_DIM + c) * 2 + 1];
        }
        tile[nn][c] = v;
    }
    __syncthreads();
#pragma unroll
    for (int it = 0; it < 16; ++it) {
        int idx = threadIdx.x + it * 256;
        int c = idx >> 5, nn = idx & 31;
        int n = n0 + nn;
        if (n < N) {
            size_t o = ((size_t)b * C_DIM + c) * N + n;
            float f = x[o];
            f = (f - statsF[(b * C_DIM + c) * 2]) * statsF[(b * C_DIM + c) * 2 + 1];
            float v = tile[nn][c] + f;
            v = v >= 0.f ? v : a * v;
            out[o] = v;
        }
    }
}

// ---------------------------------------------------------------------------
extern "C" void kernel_launch(void* const* d_in, const int* in_sizes, int n_in,
                              void* d_out, int out_size, void* d_ws, size_t ws_size,
                              hipStream_t stream) {
    const float* features = (const float*)d_in[0];
    const int*   reduce_i = (const int*)d_in[1];
    const int*   gather_i = (const int*)d_in[2];
    const float* W1s = (const float*)d_in[3];
    const float* b1s = (const float*)d_in[4];
    const float* W1o = (const float*)d_in[5];
    const float* b1o = (const float*)d_in[6];
    const float* a1  = (const float*)d_in[7];
    const float* W2s = (const float*)d_in[8];
    const float* b2s = (const float*)d_in[9];
    const float* W2o = (const float*)d_in[10];
    const float* b2o = (const float*)d_in[11];
    const float* aL  = (const float*)d_in[12];

    const int B = 2;
    const int N = in_sizes[0] / (B * C_DIM);
    const int E = in_sizes[1];

    float* ws = (float*)d_ws;
    const size_t BNC = (size_t)B * N * C_DIM;
    float* h      = ws;                          // (B,N,C) layer output / fs accum
    float* fo     = ws + BNC;                    // (B,N,C) "other" features
    float* xcm    = ws + 2 * BNC;                // (B,C,N) normalized layer-1 out
    float* Wt     = ws + 3 * BNC;                // 4 x 128x128 transposed weights
    float* sumbuf = Wt + 4 * C_DIM * C_DIM;      // B*C*2 accumulation
    float* stats1 = sumbuf + B * C_DIM * 2;
    float* stats2 = stats1 + B * C_DIM * 2;
    float* statsF = stats2 + B * C_DIM * 2;

    const float* Wt1s = Wt;
    const float* Wt1o = Wt + 1 * C_DIM * C_DIM;
    const float* Wt2s = Wt + 2 * C_DIM * C_DIM;
    const float* Wt2o = Wt + 3 * C_DIM * C_DIM;

    transpose_w<<<4, 256, 0, stream>>>(W1s, W1o, W2s, W2o, Wt);

    dim3 ggemm((N + 63) / 64, B);
    dim3 gedge(E, B);
    const int NPB = 512;
    dim3 gstat((N + NPB - 1) / NPB, B);
    dim3 gnorm((N + 31) / 32, B);

    // ---- layer 1 ----
    gemm_dual<<<ggemm, 128, 0, stream>>>(features, Wt1s, Wt1o, b1s, b1o, h, fo, N);
    scatter_add<<<gedge, 128, 0, stream>>>(h, fo, reduce_i, gather_i, N);
    hipMemsetAsync(sumbuf, 0, (size_t)B * C_DIM * 2 * sizeof(float), stream);
    stats_nmajor<<<gstat, 128, 0, stream>>>(h, sumbuf, N, NPB);
    stats_finalize<<<1, B * C_DIM, 0, stream>>>(sumbuf, stats1, N);
    norm_transpose_prelu<<<gnorm, 256, 0, stream>>>(h, stats1, a1, xcm, N);

    // ---- layer 2 ----
    gemm_dual<<<ggemm, 128, 0, stream>>>(xcm, Wt2s, Wt2o, b2s, b2o, h, fo, N);
    scatter_add<<<gedge, 128, 0, stream>>>(h, fo, reduce_i, gather_i, N);
    hipMemsetAsync(sumbuf, 0, (size_t)B * C_DIM * 2 * sizeof(float), stream);
    stats_nmajor<<<gstat, 128, 0, stream>>>(h, sumbuf, N, NPB);
    stats_finalize<<<1, B * C_DIM, 0, stream>>>(sumbuf, stats2, N);

    // ---- residual branch stats + fused output ----
    stats_cmajor<<<dim3(C_DIM, B), 256, 0, stream>>>(features, statsF, N);
    final_fuse<<<gnorm, 256, 0, stream>>>(h, stats2, features, statsF, aL,
                                          (float*)d_out, N);
}